// OrdinalLogitsModule_4818953306384
// MI455X (gfx1250) — compile-verified
//
#include <hip/hip_runtime.h>

// out[i] = cumsum(x)[i] + sum_j log_sigmoid(-x_j)
// (identity: log_sigmoid(x) - log_sigmoid(-x) == x)
//
// Memory-bound target: 268 MB @ 23.3 TB/s => ~11.5 us floor.
// CDNA5 paths: global_load_async_to_lds_b128 staging (ASYNCcnt), padded LDS
// layout for vectorized ds_load_b128 chunk reads, wave32 shuffle scan,
// native v_exp_f32/v_log_f32 softplus (~9 VALU/element keeps VALU < mem).

#define N       8192           // row length
#define TPB     256            // 8 waves of 32
#define EPT     (N / TPB)      // 32 elements per thread
#define CPT     (EPT / 4)      // 8 float4 chunks per thread
#define NWAVES  (TPB / 32)
#define STRIDEW 36             // padded words per thread chunk (144 B, 16B-aligned)
#define PWORDS  (TPB * STRIDEW)  // 9216 words = 36 KB

__device__ __forceinline__ float softplus_fast(float t) {
  // softplus(t) = max(t,0) + ln2 * log2(1 + exp2(-|t| * log2e))
  const float LOG2E = 1.4426950408889634f;
  const float LN2   = 0.6931471805599453f;
#if __has_builtin(__builtin_amdgcn_exp2f) && __has_builtin(__builtin_amdgcn_logf)
  float e = __builtin_amdgcn_exp2f(-fabsf(t) * LOG2E);   // v_exp_f32
  float l = __builtin_amdgcn_logf(1.0f + e);             // v_log_f32 (log2)
  return fmaxf(t, 0.0f) + LN2 * l;
#else
  float e = __expf(-fabsf(t));
  return fmaxf(t, 0.0f) + __logf(1.0f + e);
#endif
}

__global__ __launch_bounds__(TPB) void ordinal_logits_kernel(
    const float* __restrict__ x, float* __restrict__ out, int rows) {
  __shared__ float s_x[PWORDS];
  __shared__ float s_wx[NWAVES];
  __shared__ float s_wsp[NWAVES];

  const int tid = threadIdx.x;
  const int row = blockIdx.x;
  if (row >= rows) return;

  const float* rowIn = x + (size_t)row * N;
  float* rowOut      = out + (size_t)row * N;

  // LDS byte offset of s_x: low 32 bits of flat shared-aperture address.
  const unsigned ldsBase = (unsigned)(unsigned long long)(&s_x[0]);

  // ---- Phase 0: coalesced async copy global -> LDS ----
  // float i -> LDS word i + 4*(i/32); chunk g covers floats [4g, 4g+4)
  // => word 4g + 4*(g>>3), byte 16g + 16*(g>>3): always 16B-aligned.
#pragma unroll
  for (int i = 0; i < CPT; ++i) {
    unsigned g       = (unsigned)tid + (unsigned)TPB * i;   // 0..2047
    unsigned gOffB   = g * 16u;                             // global byte offset
    unsigned ldsOffB = ldsBase + g * 16u + (g >> 3) * 16u;  // padded LDS bytes
    asm volatile("global_load_async_to_lds_b128 %0, %1, %2"
                 :
                 : "v"(ldsOffB), "v"(gOffB), "s"(rowIn)
                 : "memory");
  }
  asm volatile("s_wait_asynccnt 0" ::: "memory");
  __syncthreads();

  // ---- Phase 1: vectorized LDS reads + per-thread sums ----
  // Thread t owns words [36t, 36t+32); base byte 144t is 16B-aligned.
  const float4* chunk = (const float4*)&s_x[STRIDEW * tid];
  float4 v[CPT];
  float sx = 0.0f, ssp = 0.0f;
#pragma unroll
  for (int j = 0; j < CPT; ++j) {
    float4 t = chunk[j];
    v[j] = t;
    sx += (t.x + t.y) + (t.z + t.w);
    ssp += softplus_fast(t.x);
    ssp += softplus_fast(t.y);
    ssp += softplus_fast(t.z);
    ssp += softplus_fast(t.w);
  }

  // ---- Phase 2: wave32 inclusive scan of sx, wave reduce of ssp ----
  const int lane = tid & 31;
  const int wid  = tid >> 5;
  float inclX = sx, wsp = ssp;
#pragma unroll
  for (int d = 1; d < 32; d <<= 1) {
    float nx = __shfl_up(inclX, d, 32);
    float ns = __shfl_up(wsp, d, 32);
    if (lane >= d) { inclX += nx; wsp += ns; }
  }
  if (lane == 31) { s_wx[wid] = inclX; s_wsp[wid] = wsp; }
  __syncthreads();

  float waveOffX = 0.0f, totSP = 0.0f;
#pragma unroll
  for (int w = 0; w < NWAVES; ++w) {
    float twx = s_wx[w];
    float tws = s_wsp[w];
    if (w < wid) waveOffX += twx;
    totSP += tws;
  }

  float acc = waveOffX + (inclX - sx);  // exclusive prefix of x for this thread
  const float R = -totSP;               // sum_j log_sigmoid(-x_j)

  // ---- Phase 3: sequential prefix within chunk, b128 stores ----
#pragma unroll
  for (int j = 0; j < CPT; ++j) {
    float4 o;
    acc += v[j].x; o.x = acc + R;
    acc += v[j].y; o.y = acc + R;
    acc += v[j].z; o.z = acc + R;
    acc += v[j].w; o.w = acc + R;
    *(float4*)(rowOut + (size_t)EPT * tid + 4 * j) = o;
  }
}

extern "C" void kernel_launch(void* const* d_in, const int* in_sizes, int n_in,
                              void* d_out, int out_size, void* d_ws, size_t ws_size,
                              hipStream_t stream) {
  (void)n_in; (void)out_size; (void)d_ws; (void)ws_size;
  const float* x = (const float*)d_in[0];
  float* out     = (float*)d_out;
  int rows = in_sizes[0] / N;  // 4096 for the reference shapes
  if (rows <= 0) return;
  ordinal_logits_kernel<<<rows, TPB, 0, stream>>>(x, out, rows);
}